// GCN_wisdm_Attn_1898375545331
// MI455X (gfx1250) — compile-verified
//
#include <hip/hip_runtime.h>
#include <hip/hip_bf16.h>

typedef __attribute__((ext_vector_type(2))) float v2f;
typedef __attribute__((ext_vector_type(8))) float v8f;

#define N_NODES   100000
#define N_EDGES0  800000
#define N_EDGES   900000   // + self loops
#define ROW_TILES 6250     // 100000 / 16

__device__ __forceinline__ float leaky(float v) { return v > 0.0f ? v : 0.2f * v; }

__device__ __forceinline__ void atomicMaxFloat(float* addr, float val) {
    // sign-aware trick: positive floats order like ints, negative like reversed uints
    if (val >= 0.0f) atomicMax((int*)addr, __float_as_int(val));
    else             atomicMin((unsigned int*)addr, __float_as_uint(val));
}

// ---------------- init: agg = 0, emax = -inf, denom = 0 ----------------
__global__ __launch_bounds__(256) void init_ws(float* agg, float* emax, float* denom) {
    size_t i = (size_t)blockIdx.x * 256 + threadIdx.x;
    if (i < (size_t)N_NODES * 256) agg[i] = 0.0f;
    if (i < N_NODES) { emax[i] = -INFINITY; denom[i] = 0.0f; }
}

// ---------------- GEMM1: xw[N,256] = x[N,128] @ W[128,256] ----------------
// One wave computes a 16x64 tile (4 WMMA col-tiles), K in steps of 4 (f32 WMMA).
__global__ __launch_bounds__(256) void gemm1_xw(const float* __restrict__ x,
                                                const float* __restrict__ W,
                                                float* __restrict__ xw) {
    const int lane = threadIdx.x & 31;
    const int wave = threadIdx.x >> 5;
    const int rowTile = blockIdx.x * 8 + wave;
    if (rowTile >= ROW_TILES) return;
    const int row0 = rowTile * 16;
    const int col0 = blockIdx.y * 64;

    const int rA  = row0 + (lane & 15);      // A row for this lane
    const int kHi = (lane >> 4) * 2;         // lanes 16-31 hold K+2/K+3
    const int nB  = lane & 15;               // B column within tile

    v8f acc[4] = {};
    for (int k = 0; k < 128; k += 4) {
        v2f a;
        a.x = x[(size_t)rA * 128 + k + kHi];
        a.y = x[(size_t)rA * 128 + k + kHi + 1];
#pragma unroll
        for (int t = 0; t < 4; ++t) {
            v2f bm;
            bm.x = W[(size_t)(k + kHi)     * 256 + col0 + t * 16 + nB];
            bm.y = W[(size_t)(k + kHi + 1) * 256 + col0 + t * 16 + nB];
            acc[t] = __builtin_amdgcn_wmma_f32_16x16x4_f32(
                false, a, false, bm, (short)0, acc[t], false, false);
        }
    }
    const int mBase = row0 + ((lane >> 4) << 3);  // lanes 16-31 -> rows +8
#pragma unroll
    for (int t = 0; t < 4; ++t)
#pragma unroll
        for (int r = 0; r < 8; ++r)
            xw[(size_t)(mBase + r) * 256 + col0 + t * 16 + nB] = acc[t][r];
}

// ---------------- per-node attention halves ----------------
__global__ __launch_bounds__(256) void alphas_kernel(const float* __restrict__ xw,
                                                     const float* __restrict__ a_src,
                                                     const float* __restrict__ a_dst,
                                                     float* __restrict__ as,
                                                     float* __restrict__ ad) {
    const int lane = threadIdx.x & 31;
    const int node = blockIdx.x * 8 + (threadIdx.x >> 5);
    if (node >= N_NODES) return;
    const float* xr = xw + (size_t)node * 256;
    float s = 0.0f, d = 0.0f;
#pragma unroll
    for (int i = 0; i < 8; ++i) {
        float v = xr[lane + 32 * i];
        s += v * a_src[lane + 32 * i];
        d += v * a_dst[lane + 32 * i];
    }
#pragma unroll
    for (int off = 16; off; off >>= 1) {
        s += __shfl_xor(s, off, 32);
        d += __shfl_xor(d, off, 32);
    }
    if (lane == 0) { as[node] = s; ad[node] = d; }
}

__device__ __forceinline__ void edge_endpoints(const int* __restrict__ ei, int e,
                                               int& s, int& d) {
    if (e < N_EDGES0) { s = ei[e]; d = ei[N_EDGES0 + e]; }
    else              { s = e - N_EDGES0; d = e - N_EDGES0; }
}

// ---------------- edge pass 1: segment max ----------------
__global__ __launch_bounds__(256) void edge_max_kernel(const int* __restrict__ ei,
                                                       const float* __restrict__ as,
                                                       const float* __restrict__ ad,
                                                       float* __restrict__ emax) {
    int e = blockIdx.x * 256 + threadIdx.x;
    if (e >= N_EDGES) return;
    int s, d; edge_endpoints(ei, e, s, d);
    atomicMaxFloat(&emax[d], leaky(as[s] + ad[d]));
}

// ---------------- edge pass 2: exp + denom ----------------
__global__ __launch_bounds__(256) void edge_exp_kernel(const int* __restrict__ ei,
                                                       const float* __restrict__ as,
                                                       const float* __restrict__ ad,
                                                       const float* __restrict__ emax,
                                                       float* __restrict__ ebuf,
                                                       float* __restrict__ denom) {
    int e = blockIdx.x * 256 + threadIdx.x;
    if (e >= N_EDGES) return;
    int s, d; edge_endpoints(ei, e, s, d);
    float ex = __expf(leaky(as[s] + ad[d]) - emax[d]);
    ebuf[e] = ex;
    atomicAdd(&denom[d], ex);
}

// ---------------- edge pass 3: weighted scatter-add ----------------
// 4 edges per block; each thread handles 4 contiguous columns (float4 gather).
__global__ __launch_bounds__(256) void edge_agg_kernel(const int* __restrict__ ei,
                                                       const float* __restrict__ ebuf,
                                                       const float* __restrict__ denom,
                                                       const float* __restrict__ xw,
                                                       float* __restrict__ agg) {
    int e = blockIdx.x * 4 + (threadIdx.x >> 6);
    if (e >= N_EDGES) return;
    int s, d; edge_endpoints(ei, e, s, d);
    float alpha = ebuf[e] / denom[d];
    int c0 = (threadIdx.x & 63) * 4;
    float4 v = *(const float4*)(xw + (size_t)s * 256 + c0);
    float* dst = agg + (size_t)d * 256 + c0;
    atomicAdd(dst + 0, alpha * v.x);
    atomicAdd(dst + 1, alpha * v.y);
    atomicAdd(dst + 2, alpha * v.z);
    atomicAdd(dst + 3, alpha * v.w);
}

// ---------------- GEMM2: h2[N,128] = relu(relu(agg+b) @ Wfc2 + bfc2) ----------------
__global__ __launch_bounds__(256) void gemm2_kernel(const float* __restrict__ agg,
                                                    const float* __restrict__ b,
                                                    const float* __restrict__ Wfc2,
                                                    const float* __restrict__ bfc2,
                                                    float* __restrict__ h2) {
    const int lane = threadIdx.x & 31;
    const int wave = threadIdx.x >> 5;
    const int rowTile = blockIdx.x * 8 + wave;
    if (rowTile >= ROW_TILES) return;
    const int row0 = rowTile * 16;
    const int col0 = blockIdx.y * 64;

    const int rA  = row0 + (lane & 15);
    const int kHi = (lane >> 4) * 2;
    const int nB  = lane & 15;

    v8f acc[4] = {};
    for (int k = 0; k < 256; k += 4) {
        v2f a;
        a.x = fmaxf(agg[(size_t)rA * 256 + k + kHi]     + b[k + kHi],     0.0f);
        a.y = fmaxf(agg[(size_t)rA * 256 + k + kHi + 1] + b[k + kHi + 1], 0.0f);
#pragma unroll
        for (int t = 0; t < 4; ++t) {
            v2f bm;
            bm.x = Wfc2[(size_t)(k + kHi)     * 128 + col0 + t * 16 + nB];
            bm.y = Wfc2[(size_t)(k + kHi + 1) * 128 + col0 + t * 16 + nB];
            acc[t] = __builtin_amdgcn_wmma_f32_16x16x4_f32(
                false, a, false, bm, (short)0, acc[t], false, false);
        }
    }
    const int mBase = row0 + ((lane >> 4) << 3);
#pragma unroll
    for (int t = 0; t < 4; ++t) {
        int col = col0 + t * 16 + nB;
#pragma unroll
        for (int r = 0; r < 8; ++r)
            h2[(size_t)(mBase + r) * 128 + col] = fmaxf(acc[t][r] + bfc2[col], 0.0f);
    }
}

// ---------------- GEMM3: out[N,6] = h2[N,128] @ Wout[128,6] + bout ----------------
__global__ __launch_bounds__(256) void gemm3_kernel(const float* __restrict__ h2,
                                                    const float* __restrict__ Wout,
                                                    const float* __restrict__ bout,
                                                    float* __restrict__ out) {
    const int lane = threadIdx.x & 31;
    const int wave = threadIdx.x >> 5;
    const int rowTile = blockIdx.x * 8 + wave;
    if (rowTile >= ROW_TILES) return;
    const int row0 = rowTile * 16;

    const int rA  = row0 + (lane & 15);
    const int kHi = (lane >> 4) * 2;
    const int nB  = lane & 15;

    v8f acc = {};
    for (int k = 0; k < 128; k += 4) {
        v2f a;
        a.x = h2[(size_t)rA * 128 + k + kHi];
        a.y = h2[(size_t)rA * 128 + k + kHi + 1];
        v2f bm;
        bm.x = (nB < 6) ? Wout[(size_t)(k + kHi)     * 6 + nB] : 0.0f;
        bm.y = (nB < 6) ? Wout[(size_t)(k + kHi + 1) * 6 + nB] : 0.0f;
        acc = __builtin_amdgcn_wmma_f32_16x16x4_f32(
            false, a, false, bm, (short)0, acc, false, false);
    }
    if (nB < 6) {
        const int mBase = row0 + ((lane >> 4) << 3);
        float bb = bout[nB];
#pragma unroll
        for (int r = 0; r < 8; ++r)
            out[(size_t)(mBase + r) * 6 + nB] = acc[r] + bb;
    }
}

extern "C" void kernel_launch(void* const* d_in, const int* in_sizes, int n_in,
                              void* d_out, int out_size, void* d_ws, size_t ws_size,
                              hipStream_t stream) {
    const float* x     = (const float*)d_in[0];
    const int*   ei    = (const int*)  d_in[1];
    const float* W     = (const float*)d_in[2];
    const float* a_src = (const float*)d_in[3];
    const float* a_dst = (const float*)d_in[4];
    const float* b     = (const float*)d_in[5];
    const float* Wfc2  = (const float*)d_in[6];
    const float* bfc2  = (const float*)d_in[7];
    const float* Wout  = (const float*)d_in[8];
    const float* bout  = (const float*)d_in[9];
    float* out = (float*)d_out;

    // workspace layout (floats)
    float* ws      = (float*)d_ws;
    float* xw      = ws;                              // N*256
    float* agg     = xw + (size_t)N_NODES * 256;      // N*256
    float* alpha_s = agg + (size_t)N_NODES * 256;     // N
    float* alpha_d = alpha_s + N_NODES;               // N
    float* emax    = alpha_d + N_NODES;               // N
    float* denom   = emax + N_NODES;                  // N
    float* ebuf    = denom + N_NODES;                 // E
    float* h2      = xw;                              // alias: xw dead after aggregation

    // 1. init
    init_ws<<<N_NODES, 256, 0, stream>>>(agg, emax, denom);
    // 2. xw = x @ W   (WMMA f32)
    gemm1_xw<<<dim3((ROW_TILES + 7) / 8, 4), 256, 0, stream>>>(x, W, xw);
    // 3. per-node attention halves
    alphas_kernel<<<(N_NODES + 7) / 8, 256, 0, stream>>>(xw, a_src, a_dst, alpha_s, alpha_d);
    // 4. edge softmax (max, exp+sum) and aggregation
    edge_max_kernel<<<(N_EDGES + 255) / 256, 256, 0, stream>>>(ei, alpha_s, alpha_d, emax);
    edge_exp_kernel<<<(N_EDGES + 255) / 256, 256, 0, stream>>>(ei, alpha_s, alpha_d, emax, ebuf, denom);
    edge_agg_kernel<<<(N_EDGES + 3) / 4, 256, 0, stream>>>(ei, ebuf, denom, xw, agg);
    // 5. fc2 + relu (WMMA f32, fused bias/relu)
    gemm2_kernel<<<dim3((ROW_TILES + 7) / 8, 2), 256, 0, stream>>>(agg, b, Wfc2, bfc2, h2);
    // 6. output head (WMMA f32)
    gemm3_kernel<<<(ROW_TILES + 7) / 8, 256, 0, stream>>>(h2, Wout, bout, out);
}